// STGCN_40922448396498
// MI455X (gfx1250) — compile-verified
//
#include <hip/hip_runtime.h>
#include <math.h>

typedef __attribute__((ext_vector_type(2))) float v2f;
typedef __attribute__((ext_vector_type(8))) float v8f;

#define NN 10000
#define EE 320000
#define T0S 10
#define T2S 8

// ---------------------------------------------------------------------------
// Generic 64-wide WMMA GEMM:  C[M,64] = act( alpha*A[M,K]@B[K,64] + bias + addend (+ C) )
// B (K<=128 rows x 64 cols, row-major) is staged in LDS once per block, in a
// swizzled pair layout so each lane's WMMA B fragment is ONE aligned ds_load_b64:
//   Bs[((kg*2 + half)*64 + col)*2 + lo] = B[4*kg + 2*half + lo][col]
// Block = 128 threads = 4 waves; each wave computes a 16x64 strip (4 wmma tiles).
// Uses V_WMMA_F32_16X16X4_F32 (fp32 in/out -> preserves reference numerics).
// A fragment: lanes 0-15 need (k0,k0+1), lanes 16-31 need (k0+2,k0+3) -> one b64 load.
// ---------------------------------------------------------------------------
__global__ __launch_bounds__(128) void gemm64_wmma(
    const float* __restrict__ A, int lda, int K,
    const float* __restrict__ B,
    float* __restrict__ C, int M,
    const float* __restrict__ bias,
    const float* __restrict__ addend,
    int accumulate, int relu_act, float alpha)
{
    __shared__ float Bs[128 * 64];
    for (int i = threadIdx.x; i < K * 64; i += blockDim.x) {
        int kr  = i >> 6, col = i & 63;
        int kg  = kr >> 2, sub = kr & 3;
        int hf  = sub >> 1, lo = sub & 1;
        Bs[((kg * 2 + hf) * 64 + col) * 2 + lo] = B[i];
    }
    __syncthreads();

    const int tid  = threadIdx.x;
    const int wave = tid >> 5;
    const int lane = tid & 31;
    const int half = lane >> 4;     // 0: lanes 0-15, 1: lanes 16-31
    const int ml   = lane & 15;
    const int r0   = blockIdx.x * 64 + wave * 16;

    int arow = r0 + ml;
    if (arow >= M) arow = M - 1;    // clamp (stores are masked; EXEC stays full for WMMA)
    // each lane's two A elements are contiguous: base + k0 + 2*half
    const float* __restrict__ Ar = A + (long)arow * lda + 2 * half;
    const float* __restrict__ Bp = &Bs[half * 128 + ml * 2];

    v8f acc[4] = {};
    for (int k0 = 0; k0 < K; k0 += 4) {
        v2f a = *(const v2f*)(Ar + k0);                  // one global_load_b64
        const float* Bk = Bp + (k0 >> 2) * 256;
#pragma unroll
        for (int c = 0; c < 4; ++c) {
            v2f b = *(const v2f*)(Bk + c * 32);          // one ds_load_b64
            acc[c] = __builtin_amdgcn_wmma_f32_16x16x4_f32(
                false, a, false, b, (short)0, acc[c], false, false);
        }
    }

#pragma unroll
    for (int vi = 0; vi < 8; ++vi) {
        int rr = r0 + half * 8 + vi;          // C/D layout: vgpr vi -> row vi | vi+8
        bool ok = (rr < M);
        long rb = (long)rr * 64;
#pragma unroll
        for (int c = 0; c < 4; ++c) {
            int colid = c * 16 + ml;
            float val = alpha * acc[c][vi];
            if (bias)              val += bias[colid];
            if (addend && ok)      val += addend[rb + colid];
            if (accumulate && ok)  val += C[rb + colid];
            if (relu_act)          val = fmaxf(val, 0.0f);
            if (ok) C[rb + colid] = val;
        }
    }
}

// ---------------------------------------------------------------------------
// Small helper kernels
// ---------------------------------------------------------------------------
__global__ __launch_bounds__(256) void fill_zero_f(float* __restrict__ p, int n) {
    int i = blockIdx.x * blockDim.x + threadIdx.x;
    if (i < n) p[i] = 0.0f;
}

__global__ __launch_bounds__(256) void count_deg_k(const int* __restrict__ row,
                                                   float* __restrict__ deg, int E) {
    int e = blockIdx.x * blockDim.x + threadIdx.x;
    if (e < E) unsafeAtomicAdd(&deg[row[e]], 1.0f);
}

__global__ __launch_bounds__(256) void finish_dinv_k(float* __restrict__ deg, int n) {
    int i = blockIdx.x * blockDim.x + threadIdx.x;
    if (i < n) {
        float d = deg[i];
        deg[i] = (d > 0.0f) ? rsqrtf(d) : 0.0f;
    }
}

// prop(v): out[col] += -(dinv[row]*dinv[col]) * v[row]   (64 feats, 16 lanes/edge, float4)
__global__ __launch_bounds__(256) void prop_scatter_k(
    const int* __restrict__ row, const int* __restrict__ col,
    const float* __restrict__ dinv,
    const float* __restrict__ v, float* __restrict__ out, int E)
{
    int tid = blockIdx.x * blockDim.x + threadIdx.x;
    if (tid >= E * 16) return;
    int e = tid >> 4;
    int g = tid & 15;
    int r = row[e], c = col[e];
    float w = -dinv[r] * dinv[c];
    const float4* vp = (const float4*)(v + (long)r * 64) + g;
    float4 x = *vp;
    float* op = out + (long)c * 64 + g * 4;
    unsafeAtomicAdd(op + 0, w * x.x);
    unsafeAtomicAdd(op + 1, w * x.y);
    unsafeAtomicAdd(op + 2, w * x.z);
    unsafeAtomicAdd(op + 3, w * x.w);
}

__device__ __forceinline__ float sigm(float q) { return 1.0f / (1.0f + __expf(-q)); }

// H = relu(P * sigmoid(Q) + R)
__global__ __launch_bounds__(256) void gate1_k(const float* __restrict__ P,
                                               const float* __restrict__ Q,
                                               const float* __restrict__ R,
                                               float* __restrict__ H, int n) {
    int i = blockIdx.x * blockDim.x + threadIdx.x;
    if (i < n) H[i] = fmaxf(P[i] * sigm(Q[i]) + R[i], 0.0f);
}

// out[n, t2, o] = relu(P * sigmoid(Q) + R)   (fused final transpose)
__global__ __launch_bounds__(256) void gate2_out_k(const float* __restrict__ P,
                                                   const float* __restrict__ Q,
                                                   const float* __restrict__ R,
                                                   float* __restrict__ out,
                                                   int Nn, int t2, int T2) {
    int i = blockIdx.x * blockDim.x + threadIdx.x;
    if (i >= Nn * 64) return;
    int n = i >> 6, o = i & 63;
    out[((long)n * T2 + t2) * 64 + o] = fmaxf(P[i] * sigm(Q[i]) + R[i], 0.0f);
}

// Weight prep: transpose [O,I,1,KS] conv weights to [I,O] panels; sum taps for tconv1
// (time-broadcast input makes tconv1 a single effective GEMM); Wb = cheb W0 - W2.
__global__ __launch_bounds__(256) void prep_weights_k(
    const float* __restrict__ w11, const float* __restrict__ w12, const float* __restrict__ w13,
    const float* __restrict__ chebw,
    const float* __restrict__ w21, const float* __restrict__ w22, const float* __restrict__ w23,
    float* __restrict__ Wt1, float* __restrict__ Wb, float* __restrict__ Wt2)
{
    int idx = blockIdx.x * blockDim.x + threadIdx.x;
    if (idx >= 4096) return;
    int i = idx >> 6, o = idx & 63;
    int s = (o * 64 + i) * 3;
    float a0 = 0.f, a1 = 0.f, a2 = 0.f;
    for (int k = 0; k < 3; ++k) { a0 += w11[s + k]; a1 += w12[s + k]; a2 += w13[s + k]; }
    Wt1[idx] = a0; Wt1[4096 + idx] = a1; Wt1[8192 + idx] = a2;
    Wb[idx] = chebw[idx] - chebw[8192 + idx];
    for (int k = 0; k < 3; ++k) {
        Wt2[k * 4096 + idx]       = w21[s + k];
        Wt2[(3 + k) * 4096 + idx] = w22[s + k];
        Wt2[(6 + k) * 4096 + idx] = w23[s + k];
    }
}

// ---------------------------------------------------------------------------
extern "C" void kernel_launch(void* const* d_in, const int* in_sizes, int n_in,
                              void* d_out, int out_size, void* d_ws, size_t ws_size,
                              hipStream_t stream) {
    const float* x      = (const float*)d_in[0];
    const int*   eidx   = (const int*)  d_in[1];
    const float* lin_w  = (const float*)d_in[2];
    const float* lin_b  = (const float*)d_in[3];
    const float* t1w1   = (const float*)d_in[4];  const float* t1b1 = (const float*)d_in[5];
    const float* t1w2   = (const float*)d_in[6];  const float* t1b2 = (const float*)d_in[7];
    const float* t1w3   = (const float*)d_in[8];  const float* t1b3 = (const float*)d_in[9];
    const float* chebw  = (const float*)d_in[10]; const float* chebb = (const float*)d_in[11];
    const float* t2w1   = (const float*)d_in[12]; const float* t2b1 = (const float*)d_in[13];
    const float* t2w2   = (const float*)d_in[14]; const float* t2b2 = (const float*)d_in[15];
    const float* t2w3   = (const float*)d_in[16]; const float* t2b3 = (const float*)d_in[17];
    float* out = (float*)d_out;

    const int N = NN, E = EE;
    const long NC = (long)N * 64;
    float* ws   = (float*)d_ws;
    float* h    = ws;
    float* bP   = ws + 1 * NC;
    float* bQ   = ws + 2 * NC;
    float* bR   = ws + 3 * NC;
    float* H1c  = ws + 4 * NC;
    float* base = ws + 5 * NC;
    float* Tx1  = ws + 6 * NC;
    float* P2   = ws + 7 * NC;
    float* G    = ws + 8 * NC;              // T0S * NC
    float* deg  = ws + 18 * NC;             // N floats (also holds dinv in-place)
    float* Wt1  = ws + 18 * NC + 10240;     // 3 * 4096
    float* Wb   = Wt1 + 3 * 4096;           // 4096
    float* Wt2  = Wb + 4096;                // 9 * 4096

    const int gGemm = (N + 63) / 64;
    const int gNC   = (int)((NC + 255) / 256);
    const int gN    = (N + 255) / 256;
    const int gE    = (E + 255) / 256;
    const int gE16  = (E * 16 + 255) / 256;

    auto gemm = [&](const float* A, int lda, int K, const float* B, float* C,
                    const float* bias, const float* addend, int acc, int act, float alpha) {
        gemm64_wmma<<<gGemm, 128, 0, stream>>>(A, lda, K, B, C, N, bias, addend, acc, act, alpha);
    };

    // Weight prep
    prep_weights_k<<<16, 256, 0, stream>>>(t1w1, t1w2, t1w3, chebw, t2w1, t2w2, t2w3,
                                           Wt1, Wb, Wt2);

    // Stage 1: h = x @ lin_w + lin_b
    gemm(x, 128, 128, lin_w, h, lin_b, nullptr, 0, 0, 1.0f);

    // Stage 2: tconv1 collapses (input constant in time) -> single gated pass
    gemm(h, 64, 64, Wt1,        bP, t1b1, nullptr, 0, 0, 1.0f);
    gemm(h, 64, 64, Wt1 + 4096, bQ, t1b2, nullptr, 0, 0, 1.0f);
    gemm(h, 64, 64, Wt1 + 8192, bR, t1b3, nullptr, 0, 0, 1.0f);
    gate1_k<<<gNC, 256, 0, stream>>>(bP, bQ, bR, H1c, (int)NC);

    // Stage 3 precompute: base = H1c @ (W0 - W2) + cheb_b   (t-invariant)
    gemm(H1c, 64, 64, Wb, base, chebb, nullptr, 0, 0, 1.0f);

    // Per-timestep ChebConv: G_t = relu(base + Tx1@W1 + 2*prop(Tx1)@W2)
    for (int t = 0; t < T0S; ++t) {
        const int* row = eidx + (long)t * 2 * E;
        const int* col = row + E;
        fill_zero_f<<<gN, 256, 0, stream>>>(deg, N);
        count_deg_k<<<gE, 256, 0, stream>>>(row, deg, E);
        finish_dinv_k<<<gN, 256, 0, stream>>>(deg, N);

        fill_zero_f<<<gNC, 256, 0, stream>>>(Tx1, (int)NC);
        prop_scatter_k<<<gE16, 256, 0, stream>>>(row, col, deg, H1c, Tx1, E);
        fill_zero_f<<<gNC, 256, 0, stream>>>(P2, (int)NC);
        prop_scatter_k<<<gE16, 256, 0, stream>>>(row, col, deg, Tx1, P2, E);

        float* Gt = G + (long)t * NC;
        gemm(Tx1, 64, 64, chebw + 4096, Gt, nullptr, base, 0, 0, 1.0f);
        gemm(P2,  64, 64, chebw + 8192, Gt, nullptr, nullptr, 1, 1, 2.0f);
    }

    // Stage 4: tconv2 (real temporal conv over 10 steps) + fused transpose to [N, 8, 64]
    const float* biases[3] = { t2b1, t2b2, t2b3 };
    float* bufs[3] = { bP, bQ, bR };
    for (int t2 = 0; t2 < T2S; ++t2) {
        for (int j = 0; j < 3; ++j) {
            gemm(G + (long)t2 * NC,       64, 64, Wt2 + (j * 3 + 0) * 4096, bufs[j],
                 biases[j], nullptr, 0, 0, 1.0f);
            gemm(G + (long)(t2 + 1) * NC, 64, 64, Wt2 + (j * 3 + 1) * 4096, bufs[j],
                 nullptr, nullptr, 1, 0, 1.0f);
            gemm(G + (long)(t2 + 2) * NC, 64, 64, Wt2 + (j * 3 + 2) * 4096, bufs[j],
                 nullptr, nullptr, 1, 0, 1.0f);
        }
        gate2_out_k<<<gNC, 256, 0, stream>>>(bP, bQ, bR, out, N, t2, T2S);
    }
}